// SelectiveScanFnWrapper_77197742178457
// MI455X (gfx1250) — compile-verified
//
#include <hip/hip_runtime.h>

#define NBATCH 2
#define NDIM   768
#define NLEN   2048
#define NST    16
#define CHUNK  16
#define NC     (NLEN / CHUNK)            /* 128 chunks per (b,d) */
#define TOTCH  (NBATCH * NDIM * NC)      /* 196608 chunks */
#define WSN    ((size_t)TOTCH * NST)     /* floats per ws array */
#define CGRP   8                         /* chunks (waves) per block in K3 */

typedef float v2f __attribute__((ext_vector_type(2)));
typedef float v8f __attribute__((ext_vector_type(8)));

__device__ __forceinline__ float softplus_f(float x) {
    return (x > 20.0f) ? x : __logf(1.0f + __expf(x));
}

// ds_swizzle group-of-32 immediate: offset = xor<<10 | or<<5 | and (and=0x1f)
#define SWZ(v, IMM) __int_as_float(__builtin_amdgcn_ds_swizzle(__float_as_int(v), (IMM)))
#define SWZ_ADD(v, IMM) (v) += SWZ(v, IMM)

// ---------------------------------------------------------------------------
// K1: per (b,d,chunk) compute chunk decay  exp(A_n * sum(delta))  and the
// state produced by the chunk starting from x=0 (16-step local scan).
// ---------------------------------------------------------------------------
__global__ __launch_bounds__(256) void k1_chunk_state(
    const float* __restrict__ u, const float* __restrict__ delta,
    const float* __restrict__ A, const float* __restrict__ B,
    const float* __restrict__ dbias,
    float* __restrict__ wsDecay, float* __restrict__ wsS)
{
    int t = blockIdx.x * 256 + threadIdx.x;        // t in [0, TOTCH)
    int c = t % NC;
    int d = (t / NC) % NDIM;
    int b = t / (NC * NDIM);
    int cs = c * CHUNK;

    const float* dp = delta + (size_t)(b * NDIM + d) * NLEN + cs;
    const float* up = u     + (size_t)(b * NDIM + d) * NLEN + cs;

    float An[NST];
#pragma unroll
    for (int n = 0; n < NST; n++) An[n] = A[d * NST + n];
    float bias = dbias[d];

    float x[NST];
#pragma unroll
    for (int n = 0; n < NST; n++) x[n] = 0.0f;
    float R = 0.0f;

    for (int tt = 0; tt < CHUNK; tt++) {
        float sp = softplus_f(dp[tt] + bias);
        float du = sp * up[tt];
        R += sp;
#pragma unroll
        for (int n = 0; n < NST; n++) {
            float a  = __expf(An[n] * sp);
            float bv = B[(size_t)(b * NST + n) * NLEN + cs + tt];
            x[n] = a * x[n] + du * bv;
        }
    }
    size_t o = ((size_t)(b * NDIM + d) * NC + c) * NST;
#pragma unroll
    for (int n = 0; n < NST; n++) {
        wsS[o + n]     = x[n];
        wsDecay[o + n] = __expf(An[n] * R);
    }
}

// ---------------------------------------------------------------------------
// K2: inter-chunk scan per (b,d,n). Overwrites wsDecay[chunk] with the
// INCOMING state x_in for that chunk (in-place after reading the decay).
// ---------------------------------------------------------------------------
__global__ __launch_bounds__(256) void k2_scan(
    float* __restrict__ wsDecay, const float* __restrict__ wsS)
{
    int i = blockIdx.x * 256 + threadIdx.x;        // i in [0, B*D*N)
    int n = i & (NST - 1);
    int d = (i >> 4) % NDIM;
    int b = i / (NST * NDIM);
    size_t base = ((size_t)(b * NDIM + d) * NC) * NST + n;

    float x = 0.0f;
    for (int c = 0; c < NC; c++) {
        size_t idx = base + (size_t)c * NST;
        float dec = wsDecay[idx];
        float s   = wsS[idx];
        wsDecay[idx] = x;          // x_in for chunk c
        x = dec * x + s;
    }
}

// ---------------------------------------------------------------------------
// K3: one wave32 per (b,d,chunk); 8 chunks of the SAME (b,d) per block so
// d-indexed loads (A, dbias, Dp) are block-uniform -> scalar path.
//   P[t,n] = C[t,n]*exp(A_n r_t),  Q[s,n] = B[s,n]*delta_s*u_s*exp(-A_n r_s)
//   y_t = rowsum_{s<=t} (P Q^T)[t,s] + sum_n P[t,n]*x_in[n]
//   out = (y + u*D) * silu(z)
// ---------------------------------------------------------------------------
__global__ __launch_bounds__(256) void k3_output(
    const float* __restrict__ u, const float* __restrict__ delta,
    const float* __restrict__ A, const float* __restrict__ B,
    const float* __restrict__ C, const float* __restrict__ Dp,
    const float* __restrict__ z, const float* __restrict__ dbias,
    const float* __restrict__ wsXin, float* __restrict__ out)
{
    __shared__ float Pl[CGRP][16][17];   // stride 17: bank-conflict-free
    __shared__ float Ql[CGRP][16][17];
    __shared__ float dl[CGRP][16];
    __shared__ float ul[CGRP][16];
    __shared__ float xl[CGRP][16];

    int w    = threadIdx.x >> 5;
    int lane = threadIdx.x & 31;
    int m    = lane & 15;
    bool lo  = lane < 16;

    // block-uniform (b,d); wave-varying chunk
    int cg = blockIdx.x % (NC / CGRP);             // chunk group
    int d  = (blockIdx.x / (NC / CGRP)) % NDIM;
    int b  = blockIdx.x / ((NC / CGRP) * NDIM);
    int c  = cg * CGRP + w;
    int cs = c * CHUNK;
    size_t base = (size_t)(b * NDIM + d) * NLEN + cs;

    if (lo) {
        dl[w][m] = softplus_f(delta[base + m] + dbias[d]);
    } else {
        ul[w][m] = u[base + m];
        xl[w][m] = wsXin[((size_t)(b * NDIM + d) * NC + c) * NST + m];
    }
    __syncthreads();

    // inclusive cumulative delta r_m (LDS broadcast reads, all lanes)
    float rr = 0.0f;
#pragma unroll
    for (int t = 0; t < 16; t++) { float v = dl[w][t]; rr += (t <= m) ? v : 0.0f; }

    float dus = dl[w][m] * ul[w][m];
    float sgn = lo ? 1.0f : -1.0f;
    float mul = lo ? 1.0f : dus;
    const float* srcb = lo ? C : B;
    float* dstb = lo ? &Pl[w][0][0] : &Ql[w][0][0];

#pragma unroll
    for (int n = 0; n < NST; n++) {
        float An = A[d * NST + n];                 // scalar load (d uniform)
        float v  = srcb[(size_t)(b * NST + n) * NLEN + cs + m];
        float e  = __expf(sgn * An * rr);
        dstb[m * 17 + n] = v * e * mul;
    }
    __syncthreads();

    // T = P * Q^T via four chained f32 WMMAs (K = 16 total).
    v8f acc = {0.f, 0.f, 0.f, 0.f, 0.f, 0.f, 0.f, 0.f};
#pragma unroll
    for (int j = 0; j < 4; j++) {
        int k0 = 4 * j + (lo ? 0 : 2);
        int k1 = 4 * j + (lo ? 1 : 3);
        v2f va; va.x = Pl[w][m][k0]; va.y = Pl[w][m][k1];
        v2f vb; vb.x = Ql[w][m][k0]; vb.y = Ql[w][m][k1];
        acc = __builtin_amdgcn_wmma_f32_16x16x4_f32(
            false, va, false, vb, (short)0, acc, false, false);
    }

    // Causal mask + row-sum within each 16-lane half via ds_swizzle xor-swaps.
    // Capture the per-row sum with a select (no predicated LDS stores):
    // after the loop, lo lane m<8 holds y_m; hi lane (16+k) holds y_{k+8}.
    float yv = 0.0f;
#pragma unroll
    for (int r = 0; r < 8; r++) {
        int t = r + (lo ? 0 : 8);
        float v = (m <= t) ? acc[r] : 0.0f;
        SWZ_ADD(v, 0x041F);   // SWAPX1
        SWZ_ADD(v, 0x081F);   // SWAPX2
        SWZ_ADD(v, 0x101F);   // SWAPX4
        SWZ_ADD(v, 0x201F);   // SWAPX8
        yv = ((m & 7) == r) ? v : yv;
    }
    // cross-half fetch: lo lane l in [8,16) takes y_l from lane l+8 (= l^24)
    float yx = SWZ(yv, 0x601F);            // xor_mask = 24
    float y  = (m < 8) ? yv : yx;

    if (lo) {
        int t = m;
#pragma unroll
        for (int n = 0; n < NST; n++) y += Pl[w][t][n] * xl[w][n];
        float uv = ul[w][t];
        float zv = z[base + t];
        float sig = __builtin_amdgcn_rcpf(1.0f + __expf(-zv));
        out[base + t] = (y + uv * Dp[d]) * (zv * sig);
    }
}

// ---------------------------------------------------------------------------
extern "C" void kernel_launch(void* const* d_in, const int* in_sizes, int n_in,
                              void* d_out, int out_size, void* d_ws, size_t ws_size,
                              hipStream_t stream) {
    const float* u     = (const float*)d_in[0];
    const float* delta = (const float*)d_in[1];
    const float* A     = (const float*)d_in[2];
    const float* B     = (const float*)d_in[3];
    const float* C     = (const float*)d_in[4];
    const float* Dp    = (const float*)d_in[5];
    const float* z     = (const float*)d_in[6];
    const float* dbias = (const float*)d_in[7];
    float* out = (float*)d_out;

    float* wsDecay = (float*)d_ws;       // later overwritten with x_in by k2
    float* wsS     = wsDecay + WSN;      // 2*WSN*4 = ~25.2 MB of d_ws used

    k1_chunk_state<<<TOTCH / 256, 256, 0, stream>>>(u, delta, A, B, dbias,
                                                    wsDecay, wsS);
    k2_scan<<<(NBATCH * NDIM * NST) / 256, 256, 0, stream>>>(wsDecay, wsS);
    k3_output<<<TOTCH / CGRP, 256, 0, stream>>>(u, delta, A, B, C, Dp, z, dbias,
                                                wsDecay, out);
}